// ModelX_13357348290920
// MI455X (gfx1250) — compile-verified
//
#include <hip/hip_runtime.h>
#include <hip/hip_bf16.h>
#include <math.h>

// GATv2 layer for MI455X (gfx1250, wave32).
// xl/xr/h (25.6MB each) are L2-resident (192MB L2); edge phase is
// L2-gather + atomic bound; GEMMs use fp32 WMMA (V_WMMA_F32_16X16X4_F32).

#define D_DIM 64
#define DOUT  128
#define NEG_SLOPE 0.2f

typedef __attribute__((ext_vector_type(2))) float v2f;
typedef __attribute__((ext_vector_type(8))) float v8f;

// monotonic float<->uint order mapping (for atomicMax-based segment max)
__device__ __forceinline__ unsigned f2ord(float f) {
    unsigned u = __float_as_uint(f);
    return (u & 0x80000000u) ? ~u : (u | 0x80000000u);
}
__device__ __forceinline__ float ord2f(unsigned u) {
    unsigned b = (u & 0x80000000u) ? (u ^ 0x80000000u) : ~u;
    return __uint_as_float(b);
}

// ---------------------------------------------------------------- init
__global__ void k_init(float* __restrict__ h, float* __restrict__ denom,
                       unsigned* __restrict__ smax, int n_nodes) {
    long long i = (long long)blockIdx.x * blockDim.x + threadIdx.x;
    if (i < (long long)n_nodes * D_DIM) h[i] = 0.0f;
    if (i < n_nodes) { denom[i] = 0.0f; smax[i] = 0x007FFFFFu; /* ord(-inf) */ }
}

// ------------------------------------------- xl = x@Wl+bl, xr = x@Wr+br
// One wave per 16-row tile. A frag (16x4 f32): lane L<16 holds
// x[r0+L][k0+{0,1}], lane L+16 holds x[r0+L-16][k0+{2,3}].
// B frag (4x16): lane L<16 holds W[k0+{0,1}][c0+L], lane L+16 W[k0+{2,3}][..].
// C/D: VGPR r -> row r + 8*(lane>=16), col = lane&15.
__global__ void k_node_linear(const float* __restrict__ x,
                              const float* __restrict__ Wl, const float* __restrict__ bl,
                              const float* __restrict__ Wr, const float* __restrict__ br,
                              float* __restrict__ xl, float* __restrict__ xr,
                              int n_nodes) {
    const int lane = threadIdx.x & 31;
    const int wave = threadIdx.x >> 5;
    const int tile = blockIdx.x * (blockDim.x >> 5) + wave;
    const int r0 = tile * 16;
    if (r0 + 16 > n_nodes) return;            // uniform per wave -> EXEC all ones
    const int lrow = lane & 15;
    const int lhi  = lane >> 4;

    v2f a[16];
    const float* xp = x + (long long)(r0 + lrow) * D_DIM + 2 * lhi;
#pragma unroll
    for (int kk = 0; kk < 16; ++kk)
        a[kk] = *(const v2f*)(xp + kk * 4);

#pragma unroll
    for (int m = 0; m < 2; ++m) {
        const float* W    = m ? Wr : Wl;
        const float* bias = m ? br : bl;
        float*       out  = m ? xr : xl;
#pragma unroll
        for (int ct = 0; ct < 4; ++ct) {
            const int col = ct * 16 + lrow;
            const float bv = bias[col];                   // bias folded into C
            v8f acc = {bv, bv, bv, bv, bv, bv, bv, bv};
#pragma unroll
            for (int kk = 0; kk < 16; ++kk) {
                const int k0 = kk * 4 + 2 * lhi;
                v2f b;
                b.x = W[(k0    ) * D_DIM + col];
                b.y = W[(k0 + 1) * D_DIM + col];
                acc = __builtin_amdgcn_wmma_f32_16x16x4_f32(
                        false, a[kk], false, b, (short)0, acc, false, false);
            }
#pragma unroll
            for (int r = 0; r < 8; ++r)
                out[(long long)(r0 + r + 8 * lhi) * D_DIM + col] = acc[r];
        }
    }
}

// ----------------- per-edge logits + segment max (one wave per edge)
__global__ void k_edge_logits(const int* __restrict__ src, const int* __restrict__ dst,
                              const float* __restrict__ xl, const float* __restrict__ xr,
                              const float* __restrict__ att,
                              float* __restrict__ logits, unsigned* __restrict__ smax,
                              int n_edges) {
    const int lane = threadIdx.x & 31;
    const long long e = ((long long)blockIdx.x * blockDim.x + threadIdx.x) >> 5;
    if (e >= n_edges) return;
    const int s = src[e], d = dst[e];
    const v2f vl = *(const v2f*)(xl + (long long)s * D_DIM + 2 * lane);
    const v2f vr = *(const v2f*)(xr + (long long)d * D_DIM + 2 * lane);
    const v2f va = *(const v2f*)(att + 2 * lane);
    float e0 = vl.x + vr.x, e1 = vl.y + vr.y;
    e0 = e0 > 0.0f ? e0 : NEG_SLOPE * e0;
    e1 = e1 > 0.0f ? e1 : NEG_SLOPE * e1;
    float p = e0 * va.x + e1 * va.y;
#pragma unroll
    for (int off = 16; off >= 1; off >>= 1) p += __shfl_xor(p, off, 32);
    if (lane == 0) {
        logits[e] = p;
        atomicMax(&smax[d], f2ord(p));
    }
}

// ----------------- decode seg-max (non-finite -> 0, per reference)
__global__ void k_decode_smax(const unsigned* __restrict__ smax,
                              float* __restrict__ smaxf, int n_nodes) {
    int i = blockIdx.x * blockDim.x + threadIdx.x;
    if (i >= n_nodes) return;
    float f = ord2f(smax[i]);
    smaxf[i] = __builtin_isfinite(f) ? f : 0.0f;
}

// ----------------- alpha = exp(logit - max[dst]); denom[dst] += alpha
__global__ void k_alpha(const float* __restrict__ logits, const int* __restrict__ dst,
                        const float* __restrict__ smaxf,
                        float* __restrict__ alpha, float* __restrict__ denom,
                        int n_edges) {
    int e = blockIdx.x * blockDim.x + threadIdx.x;
    if (e >= n_edges) return;
    const int d = dst[e];
    const float a = __expf(logits[e] - smaxf[d]);
    alpha[e] = a;
    atomicAdd(&denom[d], a);
}

// ----------------- h[dst] += (alpha/denom) * xl[src]   (wave per edge)
__global__ void k_scatter(const int* __restrict__ src, const int* __restrict__ dst,
                          const float* __restrict__ xl,
                          const float* __restrict__ alpha, const float* __restrict__ denom,
                          float* __restrict__ h, int n_edges) {
    const long long tid = (long long)blockIdx.x * blockDim.x + threadIdx.x;
    const long long e = tid >> 5;
    const int lane = (int)(tid & 31);
    if (e >= n_edges) return;
    const int s = src[e], d = dst[e];
    const float w = alpha[e] / (denom[d] + 1e-16f);
    const v2f v = *(const v2f*)(xl + (long long)s * D_DIM + 2 * lane);
    float* hp = h + (long long)d * D_DIM + 2 * lane;
    atomicAdd(hp,     w * v.x);
    atomicAdd(hp + 1, w * v.y);
}

// ----------------- out = (h[mask]+gnn_bias) @ W_out + b_out  (WMMA)
__global__ void k_output(const float* __restrict__ h, const int* __restrict__ mask,
                         const float* __restrict__ gbias,
                         const float* __restrict__ Wo, const float* __restrict__ bo,
                         float* __restrict__ out, int n_mask) {
    const int lane = threadIdx.x & 31;
    const int wave = threadIdx.x >> 5;
    const int tile = blockIdx.x * (blockDim.x >> 5) + wave;
    const int r0 = tile * 16;
    if (r0 + 16 > n_mask) return;             // uniform per wave
    const int lrow = lane & 15;
    const int lhi  = lane >> 4;

    const long long mi = (long long)mask[r0 + lrow];
    v2f a[16];
    const float* hp = h + mi * D_DIM + 2 * lhi;
    const float* gp = gbias + 2 * lhi;
#pragma unroll
    for (int kk = 0; kk < 16; ++kk) {
        const v2f hv = *(const v2f*)(hp + kk * 4);
        const v2f gv = *(const v2f*)(gp + kk * 4);
        a[kk].x = hv.x + gv.x;
        a[kk].y = hv.y + gv.y;
    }
#pragma unroll
    for (int ct = 0; ct < 8; ++ct) {
        const int col = ct * 16 + lrow;
        const float bv = bo[col];
        v8f acc = {bv, bv, bv, bv, bv, bv, bv, bv};
#pragma unroll
        for (int kk = 0; kk < 16; ++kk) {
            const int k0 = kk * 4 + 2 * lhi;
            v2f b;
            b.x = Wo[(k0    ) * DOUT + col];
            b.y = Wo[(k0 + 1) * DOUT + col];
            acc = __builtin_amdgcn_wmma_f32_16x16x4_f32(
                    false, a[kk], false, b, (short)0, acc, false, false);
        }
#pragma unroll
        for (int r = 0; r < 8; ++r)
            out[(long long)(r0 + r + 8 * lhi) * DOUT + col] = acc[r];
    }
}

extern "C" void kernel_launch(void* const* d_in, const int* in_sizes, int n_in,
                              void* d_out, int out_size, void* d_ws, size_t ws_size,
                              hipStream_t stream) {
    const float* x     = (const float*)d_in[0];
    const int*   ei    = (const int*)d_in[1];
    const int*   mask  = (const int*)d_in[2];
    const float* Wl    = (const float*)d_in[3];
    const float* bl    = (const float*)d_in[4];
    const float* Wr    = (const float*)d_in[5];
    const float* br    = (const float*)d_in[6];
    const float* att   = (const float*)d_in[7];
    const float* gbias = (const float*)d_in[8];
    const float* Wo    = (const float*)d_in[9];
    const float* bo    = (const float*)d_in[10];
    float* out = (float*)d_out;

    const int n_nodes = in_sizes[0] / D_DIM;
    const int n_edges = in_sizes[1] / 2;
    const int n_mask  = in_sizes[2];
    const int* src = ei;
    const int* dst = ei + n_edges;

    // workspace carve-up (256B aligned): xl, xr, h, logits, alpha, smax, smaxf, denom
    char* wp = (char*)d_ws;
    auto carve = [&](size_t bytes) {
        char* p = wp;
        wp += (bytes + 255) & ~(size_t)255;
        return p;
    };
    float*    xl     = (float*)   carve((size_t)n_nodes * D_DIM * 4);
    float*    xr     = (float*)   carve((size_t)n_nodes * D_DIM * 4);
    float*    h      = (float*)   carve((size_t)n_nodes * D_DIM * 4);
    float*    logits = (float*)   carve((size_t)n_edges * 4);
    float*    alpha  = (float*)   carve((size_t)n_edges * 4);
    unsigned* smax   = (unsigned*)carve((size_t)n_nodes * 4);
    float*    smaxf  = (float*)   carve((size_t)n_nodes * 4);
    float*    denom  = (float*)   carve((size_t)n_nodes * 4);

    // 1) init accumulators
    {
        long long t = (long long)n_nodes * D_DIM;
        int blocks = (int)((t + 255) / 256);
        k_init<<<blocks, 256, 0, stream>>>(h, denom, smax, n_nodes);
    }
    // 2) node linear transforms (WMMA fp32)
    {
        int tiles  = (n_nodes + 15) / 16;
        int blocks = (tiles + 7) / 8;          // 8 waves / 256-thread block
        k_node_linear<<<blocks, 256, 0, stream>>>(x, Wl, bl, Wr, br, xl, xr, n_nodes);
    }
    // 3) edge logits + segment max (wave per edge)
    {
        long long t = (long long)n_edges * 32;
        int blocks = (int)((t + 255) / 256);
        k_edge_logits<<<blocks, 256, 0, stream>>>(src, dst, xl, xr, att, logits, smax, n_edges);
    }
    // 4) decode segment max
    k_decode_smax<<<(n_nodes + 255) / 256, 256, 0, stream>>>(smax, smaxf, n_nodes);
    // 5) alpha numerators + denom
    k_alpha<<<(n_edges + 255) / 256, 256, 0, stream>>>(logits, dst, smaxf, alpha, denom, n_edges);
    // 6) weighted scatter-add
    {
        long long t = (long long)n_edges * 32;
        int blocks = (int)((t + 255) / 256);
        k_scatter<<<blocks, 256, 0, stream>>>(src, dst, xl, alpha, denom, h, n_edges);
    }
    // 7) masked output GEMM (WMMA fp32)
    {
        int tiles  = (n_mask + 15) / 16;
        int blocks = (tiles + 7) / 8;
        k_output<<<blocks, 256, 0, stream>>>(h, mask, gbias, Wo, bo, out, n_mask);
    }
}